// RNNLM_2714419331587
// MI455X (gfx1250) — compile-verified
//
#include <hip/hip_runtime.h>
#include <hip/hip_bf16.h>

#define VOCAB  500000
#define EMB    15
#define SEQ    2048
#define NTILES (VOCAB / 16)   // 31250, exact

typedef float v2f __attribute__((ext_vector_type(2)));
typedef float v8f __attribute__((ext_vector_type(8)));

__device__ __forceinline__ float fast_tanh(float x) {
    // tanh(x) = 1 - 2/(exp(2x)+1) with v_exp_f32 + v_rcp_f32 (no IEEE div
    // sequence -- keeps the serial RNN dependency chain short).
    float e = __expf(2.0f * x);
    return 1.0f - 2.0f * __builtin_amdgcn_rcpf(e + 1.0f);
}

// ---------------------------------------------------------------------------
// Kernel 1: sequential RNN scan. One workgroup.
//   Phase 1 (parallel, 512 thr): P[t][j] = we[tok[t]] . i2h[0:15, j] -> LDS
//   Phase 2 (wave 0, lanes 0..14): 2047 serial steps entirely in regs/LDS
//   Writes combined[32] = {we[tok[last]], h_final, 0, 0} to ws[0..31].
// ---------------------------------------------------------------------------
__global__ void rnn_scan_kernel(const int* __restrict__ tok,
                                const float* __restrict__ we,
                                const float* __restrict__ hidden,
                                const float* __restrict__ i2h,
                                float* __restrict__ ws)
{
    extern __shared__ float P[];            // [SEQ*16] floats (128 KB dynamic LDS)
    __shared__ float ihs[2 * EMB * EMB];    // i2h, 30x15
    const int tid = threadIdx.x;

    for (int i = tid; i < 2 * EMB * EMB; i += blockDim.x) ihs[i] = i2h[i];
    __syncthreads();

    // Phase 1: parallel input projection (removes gather+GEMV from serial chain)
    for (int u = tid; u < (SEQ - 1) * 16; u += blockDim.x) {
        const int t = u >> 4, j = u & 15;
        float acc = 0.0f;
        if (j < EMB) {
            const float* e = we + (long)tok[t] * EMB;
            #pragma unroll
            for (int k = 0; k < EMB; ++k) acc = fmaf(e[k], ihs[k * EMB + j], acc);
        }
        P[u] = acc;
    }
    __syncthreads();

    // Phase 2: serial scan, lane j owns h[j]; cross-lane h via wave32 shuffles
    if (tid < EMB) {
        const int j = tid;
        float wh[EMB];
        #pragma unroll
        for (int i = 0; i < EMB; ++i) wh[i] = ihs[(EMB + i) * EMB + j];
        float h = hidden[j];
        for (int t = 0; t < SEQ - 1; ++t) {
            float acc = P[t * 16 + j];
            #pragma unroll
            for (int i = 0; i < EMB; ++i)
                acc = fmaf(__shfl(h, i, 32), wh[i], acc);
            h = fast_tanh(acc);
        }
        ws[j]       = we[(long)tok[SEQ - 1] * EMB + j];
        ws[EMB + j] = h;
        if (j == 0) { ws[2 * EMB] = 0.0f; ws[2 * EMB + 1] = 0.0f; }  // K-pad
    }
}

// ---------------------------------------------------------------------------
// Kernel 2: out = exp(tanh(combined @ i2o)) via V_WMMA_F32_16X16X4_F32.
// Bandwidth-bound: streams the 60 MB i2o matrix (L2-resident on replay).
// A fragments are loop-invariant and hoisted to registers; per tile all 16
// B words are loaded first (one clause), then 8 chained WMMAs with
// progressive loadcnt waits.
// ---------------------------------------------------------------------------
__global__ void i2o_wmma_kernel(const float* __restrict__ i2o,
                                const float* __restrict__ ws,
                                float* __restrict__ out,
                                float* __restrict__ partials)
{
    __shared__ float comb[32];
    __shared__ float wsum[8];
    const int tid  = threadIdx.x;
    const int lane = tid & 31;
    const int wv   = tid >> 5;
    if (tid < 32) comb[tid] = ws[tid];
    __syncthreads();

    const int m  = lane & 15;   // A: M index / B,D: N index
    const int hi = lane >> 4;   // half-wave selects K pair

    // Hoisted A fragments: row 0 = combined[k..k+3], rows 1..15 = 0.
    // Unconditional LDS reads + select -> no exec-mask churn, no per-tile DS.
    v2f afrag[8];
    long roff0[8], roff1[8];
    #pragma unroll
    for (int kk = 0; kk < 8; ++kk) {
        const int k0 = kk * 4;
        float ax = comb[k0 + 2 * hi];        // comb[30..31] == 0 (K pad)
        float ay = comb[k0 + 2 * hi + 1];
        afrag[kk].x = (m == 0) ? ax : 0.0f;
        afrag[kk].y = (m == 0) ? ay : 0.0f;
        int r0 = k0 + 2 * hi;  if (r0 > 29) r0 = 29;  // pad rows: A is 0 there
        int r1 = r0 + 1;       if (r1 > 29) r1 = 29;
        roff0[kk] = (long)r0 * VOCAB;
        roff1[kk] = (long)r1 * VOCAB;
    }

    const int gwave  = blockIdx.x * (blockDim.x >> 5) + wv;
    const int nwaves = gridDim.x * (blockDim.x >> 5);

    float lsum = 0.0f;
    for (int tile = gwave; tile < NTILES; tile += nwaves) {
        const long v0 = (long)tile * 16;
        const float* bp = i2o + v0 + m;

        // Batch all B loads for this tile (16 dwords -> one clause)
        v2f b[8];
        #pragma unroll
        for (int kk = 0; kk < 8; ++kk) {
            b[kk].x = bp[roff0[kk]];
            b[kk].y = bp[roff1[kk]];
        }

        // Back-to-back WMMA accumulation chain (K = 32, padded from 30)
        v8f c = {0.f, 0.f, 0.f, 0.f, 0.f, 0.f, 0.f, 0.f};
        #pragma unroll
        for (int kk = 0; kk < 8; ++kk) {
            c = __builtin_amdgcn_wmma_f32_16x16x4_f32(false, afrag[kk], false, b[kk],
                                                      (short)0, c, false, false);
        }

        if (lane < 16) {                 // D row M=0 lives in elem 0, lanes 0..15
            float e = __expf(fast_tanh(c[0]));
            out[v0 + lane] = e;
            lsum += e;
        }
    }
    #pragma unroll
    for (int off = 16; off > 0; off >>= 1)
        lsum += __shfl_down(lsum, off, 32);
    if (lane == 0) wsum[wv] = lsum;
    __syncthreads();
    if (tid == 0) {
        float s = 0.0f;
        #pragma unroll
        for (int i = 0; i < 8; ++i) s += wsum[i];   // fixed order -> deterministic
        partials[blockIdx.x] = s;
    }
}

// ---------------------------------------------------------------------------
// Kernel 3: deterministic reduction of block partials -> ws[32] = 1/sum
// ---------------------------------------------------------------------------
__global__ void reduce_kernel(const float* __restrict__ partials,
                              float* __restrict__ ws, int n)
{
    __shared__ float s[256];
    const int tid = threadIdx.x;
    float a = 0.0f;
    for (int i = tid; i < n; i += 256) a += partials[i];  // fixed order
    s[tid] = a;
    __syncthreads();
    for (int off = 128; off > 0; off >>= 1) {
        if (tid < off) s[tid] += s[tid + off];
        __syncthreads();
    }
    if (tid == 0) ws[32] = 1.0f / s[0];
}

// ---------------------------------------------------------------------------
// Kernel 4: out[i] *= 1/sum
// ---------------------------------------------------------------------------
__global__ void normalize_kernel(float* __restrict__ out,
                                 const float* __restrict__ ws)
{
    const int i = blockIdx.x * blockDim.x + threadIdx.x;
    const float inv = ws[32];
    if (i < VOCAB) out[i] *= inv;
}

extern "C" void kernel_launch(void* const* d_in, const int* in_sizes, int n_in,
                              void* d_out, int out_size, void* d_ws, size_t ws_size,
                              hipStream_t stream)
{
    const int*   tok    = (const int*)  d_in[0];
    const float* we     = (const float*)d_in[1];
    const float* hidden = (const float*)d_in[2];
    const float* i2h    = (const float*)d_in[3];
    const float* i2o    = (const float*)d_in[4];
    float* out = (float*)d_out;
    float* ws  = (float*)d_ws;           // [0..31] combined, [32] inv, [64..] partials
    float* partials = ws + 64;

    const int NB = 1024;                 // 8192 waves streaming i2o
    rnn_scan_kernel<<<1, 512, SEQ * 16 * sizeof(float), stream>>>(tok, we, hidden, i2h, ws);
    i2o_wmma_kernel<<<NB, 256, 0, stream>>>(i2o, ws, out, partials);
    reduce_kernel<<<1, 256, 0, stream>>>(partials, ws, NB);
    normalize_kernel<<<(VOCAB + 255) / 256, 256, 0, stream>>>(out, ws);
}